// CorrelationLayer_11235634446572
// MI455X (gfx1250) — compile-verified
//
#include <hip/hip_runtime.h>
#include <hip/hip_bf16.h>

typedef __attribute__((ext_vector_type(2))) float v2f;
typedef __attribute__((ext_vector_type(8))) float v8f;

#define B_   16
#define C_   256
#define H_   64
#define W_   128
#define CK   16                  // channels per LDS chunk
#define NCHUNK (C_ / CK)         // 16
#define F1STR 18                 // padded per-pixel channel stride (even -> 8B aligned, conflict-free)
#define F2W  136                 // f2 staged width (w-4 .. w+131)
#define F2OFF (W_ * F1STR)       // 2304 floats
#define BUF_FLOATS (F2OFF + 9 * F2W * F1STR)   // 24336 floats = 97344 B per buffer
#define LDS_FLOATS (2 * BUF_FLOATS)            // 48672 floats = 194688 B (double buffer)

#if __has_builtin(__builtin_amdgcn_global_load_async_to_lds_b32)
#define ASYNC_STAGE 1
#else
#define ASYNC_STAGE 0
#endif

typedef __attribute__((address_space(1))) int gas_int;  // global
typedef __attribute__((address_space(3))) int las_int;  // LDS

__device__ __forceinline__ void stage_cp(const float* __restrict__ g, float* l) {
#if ASYNC_STAGE
  // Direct memory->LDS async copy (ASYNCcnt); per-lane LDS dest does the transpose scatter.
  __builtin_amdgcn_global_load_async_to_lds_b32((gas_int*)g, (las_int*)l, 0, 0);
#else
  *l = *g;  // fallback: global load + ds_store
#endif
}

__device__ __forceinline__ void stage_chunk(float* __restrict__ buf,
                                            const float* __restrict__ f1,
                                            const float* __restrict__ f2,
                                            int b, int h, int c0, int tid) {
  // f1[b, c0..c0+15, h, :] transposed -> buf[w*F1STR + c]   (2048 elems, 4/thread)
#pragma unroll
  for (int s = 0; s < 4; ++s) {
    const int idx = tid + s * 512;
    const int c = idx >> 7, w = idx & 127;
    stage_cp(f1 + (((size_t)b * C_ + (c0 + c)) * H_ + h) * W_ + w,
             buf + w * F1STR + c);
  }
  // f2 9-row halo transposed -> buf[F2OFF + (dy*F2W+wi)*F1STR + c]; OOB lanes masked
  for (int idx = tid; idx < CK * 9 * F2W; idx += 512) {
    const int c  = idx / (9 * F2W);
    const int r  = idx - c * (9 * F2W);
    const int dy = r / F2W;
    const int wi = r - dy * F2W;
    const int row = h + dy - 4;
    const int col = wi - 4;
    if ((unsigned)row < (unsigned)H_ && (unsigned)col < (unsigned)W_)
      stage_cp(f2 + (((size_t)b * C_ + (c0 + c)) * H_ + row) * W_ + col,
               buf + F2OFF + ((dy * F2W) + wi) * F1STR + c);
  }
}

__device__ __forceinline__ void stage_wait() {
  asm volatile("s_wait_asynccnt 0x0" ::: "memory");
}

__global__ __launch_bounds__(512)
void corr81_wmma_kernel(const float* __restrict__ f1,
                        const float* __restrict__ f2,
                        float* __restrict__ out) {
  extern __shared__ float lds[];

  const int h    = blockIdx.x;       // 0..63
  const int b    = blockIdx.y;       // 0..15
  const int tid  = (int)threadIdx.x; // 0..511
  const int lane = tid & 31;
  const int wv   = tid >> 5;         // wave 0..15
  const int m    = lane & 15;        // fragment row/col index
  const int kb   = (lane >> 4) << 1; // K sub-base: 0 (lanes 0-15) or 2 (lanes 16-31)

  // 9 chains per wave; chain id = wv*9+q -> (dy, w-tile, half)
  int dyA[9], w0A[9], halfA[9], aB[9], bB[9];
  v8f acc[9];
  const v8f vzero = {};
#pragma unroll
  for (int q = 0; q < 9; ++q) {
    const int cid  = wv * 9 + q;     // 0..143
    const int dy   = cid >> 4;       // 0..8
    const int rem  = cid & 15;
    const int t    = rem >> 1;       // w tile 0..7
    const int half = rem & 1;        // B offset: -4 (half=0) or +4 (half=1)
    const int w0   = t << 4;
    dyA[q] = dy; w0A[q] = w0; halfA[q] = half;
    aB[q] = (w0 + m) * F1STR;
    bB[q] = F2OFF + ((dy * F2W) + (w0 + 8 * half + m)) * F1STR;
    acc[q] = vzero;
  }

  // One-time zero fill (keeps chunk-invariant OOB halo slots at zero in both buffers)
  for (int i = tid; i < LDS_FLOATS; i += 512) lds[i] = 0.0f;
  __syncthreads();

  // Prologue: stage chunk 0
  stage_chunk(lds, f1, f2, b, h, 0, tid);
  stage_wait();
  __syncthreads();

  for (int ch = 0; ch < NCHUNK; ++ch) {
    float* cur = lds + ((ch & 1) ? BUF_FLOATS : 0);
    float* nxt = lds + ((ch & 1) ? 0 : BUF_FLOATS);

    // Kick off async staging of the next chunk while we compute on `cur`
    if (ch + 1 < NCHUNK)
      stage_chunk(nxt, f1, f2, b, h, (ch + 1) * CK, tid);

    // 4 K-steps of 4 channels; batch all 18 fragment loads, then 9 WMMAs (ILP)
#pragma unroll
    for (int ks = 0; ks < 4; ++ks) {
      const int cc = (ks << 2) + kb;
      v2f afr[9], bfr[9];
#pragma unroll
      for (int q = 0; q < 9; ++q) {
        afr[q] = *(const v2f*)(cur + aB[q] + cc);
        bfr[q] = *(const v2f*)(cur + bB[q] + cc);
      }
#pragma unroll
      for (int q = 0; q < 9; ++q)
        acc[q] = __builtin_amdgcn_wmma_f32_16x16x4_f32(
            /*neg_a=*/false, afr[q], /*neg_b=*/false, bfr[q],
            /*c_mod=*/(short)0, acc[q], /*reuse_a=*/false, /*reuse_b=*/false);
    }

    stage_wait();
    __syncthreads();
  }

  // Writeout. D layout: lane holds N=j=lane&15; element r holds M=i=r+8*(lane>=16).
  // dx = 8*half + j - i. half=0 uniquely covers i+dx<=15; half=1 covers i+dx>=16 (j>=8).
  const int j   = m;
  const int hi8 = (lane >> 4) << 3;
#pragma unroll
  for (int q = 0; q < 9; ++q) {
    const int dy = dyA[q], w0 = w0A[q], half = halfA[q];
#pragma unroll
    for (int r = 0; r < 8; ++r) {
      const int i  = r + hi8;
      const int dx = (half << 3) + j - i;
      if (dx >= 0 && dx <= 8 && (half == 0 || j >= 8)) {
        const size_t o = (((size_t)b * 81 + (dy * 9 + dx)) * H_ + h) * W_ + (w0 + i);
        out[o] = acc[q][r] * (1.0f / 256.0f);
      }
    }
  }
}

extern "C" void kernel_launch(void* const* d_in, const int* in_sizes, int n_in,
                              void* d_out, int out_size, void* d_ws, size_t ws_size,
                              hipStream_t stream) {
  const float* f1 = (const float*)d_in[0];
  const float* f2 = (const float*)d_in[1];
  float* out = (float*)d_out;

  const size_t shmem = (size_t)LDS_FLOATS * sizeof(float);  // 194688 B (< 320 KB WGP LDS)
  (void)hipFuncSetAttribute((const void*)corr81_wmma_kernel,
                            hipFuncAttributeMaxDynamicSharedMemorySize, (int)shmem);
  dim3 grid(H_, B_);
  corr81_wmma_kernel<<<grid, dim3(512), shmem, stream>>>(f1, f2, out);
}